// STGCN_50878182588494
// MI455X (gfx1250) — compile-verified
//
#include <hip/hip_runtime.h>
#include <stdint.h>

// ---------------- problem constants (match reference) ----------------
#define BATCH   16
#define SDIM    12
#define NNODES  20000
#define HDIM    128
#define PDIM    3
#define LAYERS  2
#define KCHEB   3
#define LN_EPS  1e-5f
#define NPAD    (NNODES + 2)   // conv input padded with one zero row each side

typedef float v2f __attribute__((ext_vector_type(2)));
typedef float v8f __attribute__((ext_vector_type(8)));

static __device__ __forceinline__ void atomic_add_f32(float* p, float v) {
  __hip_atomic_fetch_add(p, v, __ATOMIC_RELAXED, __HIP_MEMORY_SCOPE_AGENT);
}

// ---------------- graph normalization ----------------
__global__ void k_degree(const int* __restrict__ src, float* __restrict__ deg, int E) {
  int e = blockIdx.x * blockDim.x + threadIdx.x;
  if (e < E) atomic_add_f32(&deg[src[e]], 1.0f);
}

__global__ void k_norm(const int* __restrict__ src, const int* __restrict__ dst,
                       const float* __restrict__ deg, float* __restrict__ nrm, int E) {
  int e = blockIdx.x * blockDim.x + threadIdx.x;
  if (e >= E) return;
  float ds = deg[src[e]], dd = deg[dst[e]];
  float is = ds > 0.0f ? rsqrtf(fmaxf(ds, 1.0f)) : 0.0f;
  float id = dd > 0.0f ? rsqrtf(fmaxf(dd, 1.0f)) : 0.0f;
  nrm[e] = -is * id;
}

// ---------------- input projection: h[b,n,c] = sum_s x[b,s,n]*W[s,c] + b[c] ----------------
__global__ void k_input_proj(const float* __restrict__ x, const float* __restrict__ w,
                             const float* __restrict__ bias, float* __restrict__ h, long total) {
  long i = (long)blockIdx.x * blockDim.x + threadIdx.x;
  if (i >= total) return;
  int c = (int)(i % HDIM);
  long nb = i / HDIM;
  int n  = (int)(nb % NNODES);
  int bb = (int)(nb / NNODES);
  float acc = bias[c];
  const float* xb = x + (long)bb * SDIM * NNODES + n;
#pragma unroll
  for (int s = 0; s < SDIM; ++s) acc += xb[(long)s * NNODES] * w[s * HDIM + c];
  h[i] = acc;
}

// ---------------- graph scatter: out[b,dst[e],:] += nrm[e]*in[b,src[e],:] ----------------
// one wave32 per (edge,batch); 4 channels per lane (float4 gather, 4 fp atomics)
__global__ void k_prop(const float* __restrict__ in, float* __restrict__ out,
                       const int* __restrict__ src, const int* __restrict__ dst,
                       const float* __restrict__ nrm, int E) {
  long gw = ((long)blockIdx.x * blockDim.x + threadIdx.x) >> 5;
  int lane = threadIdx.x & 31;
  long total = (long)E * BATCH;
  if (gw >= total) return;
  int e = (int)(gw % E);
  int b = (int)(gw / E);
  float w = nrm[e];
  long bi = ((long)b * NNODES + src[e]) * HDIM + lane * 4;
  long bo = ((long)b * NNODES + dst[e]) * HDIM + lane * 4;
  float4 v = *(const float4*)(in + bi);
  float* o = out + bo;
  atomic_add_f32(o + 0, w * v.x);
  atomic_add_f32(o + 1, w * v.y);
  atomic_add_f32(o + 2, w * v.z);
  atomic_add_f32(o + 3, w * v.w);
}

// ---------------- Tx2 = 2*prop(Tx1) - Tx0 fixup ----------------
__global__ void k_fixup(float* __restrict__ tx2, const float* __restrict__ tx0, long total) {
  long i = (long)blockIdx.x * blockDim.x + threadIdx.x;
  if (i < total) tx2[i] = 2.0f * tx2[i] - tx0[i];
}

// ---------------- zero the conv-input padding rows (rows -1 and N per batch) ----------------
__global__ void k_zero_pad(float* __restrict__ accP) {
  int i = blockIdx.x * blockDim.x + threadIdx.x;      // BATCH*2*HDIM = 4096 threads
  if (i >= BATCH * 2 * HDIM) return;
  int b    = i / (2 * HDIM);
  int rem  = i % (2 * HDIM);
  int side = rem / HDIM;
  int c    = rem % HDIM;
  long row = (long)b * NPAD + (side ? (NPAD - 1) : 0);
  accP[row * HDIM + c] = 0.0f;
}

// ---------------- fused Cheb GEMM: outP[interior] = Tx0@W0 + Tx1@W1 + Tx2@W2 + bias ----------------
// 8 waves/block; wave -> 16x16 output tile; K stepped by 4 via v_wmma_f32_16x16x4_f32.
// Weight slab (3 x 128 x 16 cols) staged through LDS once per block, ds_load in the K loop.
// A rows clamped (not predicated): an OOB A row only feeds its own D row, which is skipped at store.
__global__ __launch_bounds__(256) void k_cheb_gemm(
    const float* __restrict__ tx0, const float* __restrict__ tx1, const float* __restrict__ tx2,
    const float* __restrict__ W, const float* __restrict__ bias, float* __restrict__ outP) {
  __shared__ float sW[KCHEB * HDIM * 16];             // [kmat][k][col], 24 KB
  int f0 = blockIdx.y * 16;
  for (int i = threadIdx.x; i < KCHEB * HDIM * 16; i += 256) {
    int km = i / (HDIM * 16);
    int rm = i % (HDIM * 16);
    int k  = rm / 16;
    int c  = rm % 16;
    sW[i] = W[(size_t)km * HDIM * HDIM + k * HDIM + f0 + c];
  }
  __syncthreads();

  int wave = threadIdx.x >> 5, lane = threadIdx.x & 31;
  int half = lane >> 4, r = lane & 15;
  int m0 = blockIdx.x * 128 + wave * 16;
  long bbase = (long)blockIdx.z * NNODES * HDIM;
  long obase = (long)blockIdx.z * NPAD * HDIM;

  int row   = m0 + r;
  int row_l = row < NNODES ? row : (NNODES - 1);      // clamp: load always in-bounds
  long arow = bbase + (long)row_l * HDIM;

  const float* sW0 = sW;
  const float* sW1 = sW + HDIM * 16;
  const float* sW2 = sW + 2 * HDIM * 16;

  v8f acc = {};
  for (int k0 = 0; k0 < HDIM; k0 += 4) {
    int ka = k0 + half * 2;        // A: lane<16 -> K {k0,k0+1}; lane>=16 -> {k0+2,k0+3}
    v2f a0 = *(const v2f*)(tx0 + arow + ka);
    v2f a1 = *(const v2f*)(tx1 + arow + ka);
    v2f a2 = *(const v2f*)(tx2 + arow + ka);
    v2f b0, b1, b2;                // B: rows K=ka,ka+1; cols f0..f0+15 across lanes
    b0.x = sW0[ka * 16 + r]; b0.y = sW0[(ka + 1) * 16 + r];
    b1.x = sW1[ka * 16 + r]; b1.y = sW1[(ka + 1) * 16 + r];
    b2.x = sW2[ka * 16 + r]; b2.y = sW2[(ka + 1) * 16 + r];
    acc = __builtin_amdgcn_wmma_f32_16x16x4_f32(false, a0, false, b0, (short)0, acc, false, false);
    acc = __builtin_amdgcn_wmma_f32_16x16x4_f32(false, a1, false, b1, (short)0, acc, false, false);
    acc = __builtin_amdgcn_wmma_f32_16x16x4_f32(false, a2, false, b2, (short)0, acc, false, false);
  }
  float bv = bias[f0 + r];
#pragma unroll
  for (int j = 0; j < 8; ++j) {    // D VGPR j: row m0 + half*8 + j, col f0 + r
    int ro = m0 + half * 8 + j;
    if (ro < NNODES)               // write into padded interior (+1 row)
      outP[obase + (long)(ro + 1) * HDIM + f0 + r] = acc[j] + bv;
  }
}

// ---------------- temporal conv (kernel=3, SAME) as 3 shifted WMMA GEMMs ----------------
// Input is the padded cheb output (zero rows at -1 and N) -> unconditional shifted loads.
// Weights CW[f,h,kw] reordered into LDS as [kw][h][col] once per block.
__global__ __launch_bounds__(256) void k_tconv_gemm(
    const float* __restrict__ inP, const float* __restrict__ CW,
    const float* __restrict__ bias, float* __restrict__ out) {
  __shared__ float sW[3 * HDIM * 16];                 // [kw][h][col], 24 KB
  int f0 = blockIdx.y * 16;
  for (int i = threadIdx.x; i < 3 * HDIM * 16; i += 256) {
    int kw = i / (HDIM * 16);
    int rm = i % (HDIM * 16);
    int hh = rm / 16;
    int c  = rm % 16;
    sW[i] = CW[((f0 + c) * HDIM + hh) * 3 + kw];
  }
  __syncthreads();

  int wave = threadIdx.x >> 5, lane = threadIdx.x & 31;
  int half = lane >> 4, r = lane & 15;
  int m0 = blockIdx.x * 128 + wave * 16;
  long ibase = (long)blockIdx.z * NPAD * HDIM;
  long obase = (long)blockIdx.z * NNODES * HDIM;

  int row = m0 + r;
  v8f acc = {};
  for (int k0 = 0; k0 < HDIM; k0 += 4) {
    int ka = k0 + half * 2;
#pragma unroll
    for (int kw = 0; kw < 3; ++kw) {
      int rin = row + kw - 1;
      if (rin > NNODES) rin = NNODES;                 // tail-tile clamp onto zero pad row
      v2f a = *(const v2f*)(inP + ibase + (long)(rin + 1) * HDIM + ka);
      v2f bf;
      bf.x = sW[(kw * HDIM + ka) * 16 + r];
      bf.y = sW[(kw * HDIM + ka + 1) * 16 + r];
      acc = __builtin_amdgcn_wmma_f32_16x16x4_f32(false, a, false, bf, (short)0, acc, false, false);
    }
  }
  float bv = bias[f0 + r];
#pragma unroll
  for (int j = 0; j < 8; ++j) {
    int ro = m0 + half * 8 + j;
    if (ro < NNODES) out[obase + (long)ro * HDIM + f0 + r] = acc[j] + bv;
  }
}

// ---------------- h = relu(LN(conv + residual)*g + b), wave per row ----------------
__global__ __launch_bounds__(256) void k_ln_relu(
    const float* __restrict__ conv, float* __restrict__ h,
    const float* __restrict__ g, const float* __restrict__ be) {
  int wave = threadIdx.x >> 5, lane = threadIdx.x & 31;
  long rowi = (long)blockIdx.x * 8 + wave;
  if (rowi >= (long)BATCH * NNODES) return;
  long base = rowi * HDIM;
  float v[4];
  float s = 0.0f;
#pragma unroll
  for (int j = 0; j < 4; ++j) { int c = lane + j * 32; v[j] = conv[base + c] + h[base + c]; s += v[j]; }
  for (int o = 16; o > 0; o >>= 1) s += __shfl_xor(s, o, 32);
  float mu = s * (1.0f / HDIM);
  float q = 0.0f;
#pragma unroll
  for (int j = 0; j < 4; ++j) { float d = v[j] - mu; q += d * d; }
  for (int o = 16; o > 0; o >>= 1) q += __shfl_xor(q, o, 32);
  float inv = rsqrtf(q * (1.0f / HDIM) + LN_EPS);
#pragma unroll
  for (int j = 0; j < 4; ++j) {
    int c = lane + j * 32;
    float y = (v[j] - mu) * inv * g[c] + be[c];
    h[base + c] = fmaxf(y, 0.0f);
  }
}

// ---------------- output: out[b,p,n] = h[b,n,:]@out_w[:,p] + out_b[p] ----------------
__global__ __launch_bounds__(256) void k_out_proj(
    const float* __restrict__ h, const float* __restrict__ w,
    const float* __restrict__ b, float* __restrict__ out) {
  int wave = threadIdx.x >> 5, lane = threadIdx.x & 31;
  long rowi = (long)blockIdx.x * 8 + wave;
  if (rowi >= (long)BATCH * NNODES) return;
  int n  = (int)(rowi % NNODES);
  int bb = (int)(rowi / NNODES);
  long base = rowi * HDIM;
  float p0 = 0.f, p1 = 0.f, p2 = 0.f;
#pragma unroll
  for (int j = 0; j < 4; ++j) {
    int c = lane + j * 32;
    float hv = h[base + c];
    p0 += hv * w[c * PDIM + 0];
    p1 += hv * w[c * PDIM + 1];
    p2 += hv * w[c * PDIM + 2];
  }
  for (int o = 16; o > 0; o >>= 1) {
    p0 += __shfl_xor(p0, o, 32);
    p1 += __shfl_xor(p1, o, 32);
    p2 += __shfl_xor(p2, o, 32);
  }
  if (lane == 0) {
    out[((long)bb * PDIM + 0) * NNODES + n] = p0 + b[0];
    out[((long)bb * PDIM + 1) * NNODES + n] = p1 + b[1];
    out[((long)bb * PDIM + 2) * NNODES + n] = p2 + b[2];
  }
}

// ---------------- driver ----------------
extern "C" void kernel_launch(void* const* d_in, const int* in_sizes, int n_in,
                              void* d_out, int out_size, void* d_ws, size_t ws_size,
                              hipStream_t stream) {
  (void)n_in; (void)out_size; (void)ws_size;
  const float* x      = (const float*)d_in[0];
  const int*   ei     = (const int*)d_in[1];
  const float* in_w   = (const float*)d_in[2];
  const float* in_b   = (const float*)d_in[3];
  const float* cheb_w = (const float*)d_in[4];
  const float* cheb_b = (const float*)d_in[5];
  const float* conv_w = (const float*)d_in[6];
  const float* conv_b = (const float*)d_in[7];
  const float* ln_g   = (const float*)d_in[8];
  const float* ln_b   = (const float*)d_in[9];
  const float* out_w  = (const float*)d_in[10];
  const float* out_b  = (const float*)d_in[11];

  const int E = in_sizes[1] / 2;
  const int* src = ei;
  const int* dst = ei + E;

  char* ws = (char*)d_ws;
  size_t off = 0;
  auto walloc = [&](size_t bytes) -> void* {
    void* p = ws + off;
    off = (off + bytes + 255) & ~(size_t)255;
    return p;
  };
  const size_t HS  = (size_t)BATCH * NNODES * HDIM;   // hidden tensor elems
  const size_t HSP = (size_t)BATCH * NPAD * HDIM;     // padded conv-input elems
  float* deg  = (float*)walloc((size_t)NNODES * 4);
  float* nrm  = (float*)walloc((size_t)E * 4);
  float* h    = (float*)walloc(HS * 4);               // hidden / residual
  float* tx1  = (float*)walloc(HS * 4);               // Tx1, later conv output
  float* tx2  = (float*)walloc(HS * 4);               // Tx2
  float* accP = (float*)walloc(HSP * 4);              // cheb output, padded rows

  // graph norm
  hipMemsetAsync(deg, 0, (size_t)NNODES * 4, stream);
  k_degree<<<(E + 255) / 256, 256, 0, stream>>>(src, deg, E);
  k_norm<<<(E + 255) / 256, 256, 0, stream>>>(src, dst, deg, nrm, E);

  // zero conv-input padding rows (cheb_gemm only ever writes the interior)
  k_zero_pad<<<(BATCH * 2 * HDIM + 255) / 256, 256, 0, stream>>>(accP);

  // input projection
  const long total = (long)HS;
  k_input_proj<<<(unsigned)((total + 255) / 256), 256, 0, stream>>>(x, in_w, in_b, h, total);

  dim3 gemm_grid((NNODES + 127) / 128, HDIM / 16, BATCH);
  const long waves = (long)E * BATCH;
  const unsigned prop_blocks = (unsigned)((waves + 7) / 8);   // 8 waves per 256-thread block
  const unsigned row_blocks  = (unsigned)(((long)BATCH * NNODES) / 8);

  for (int l = 0; l < LAYERS; ++l) {
    hipMemsetAsync(tx1, 0, HS * 4, stream);
    k_prop<<<prop_blocks, 256, 0, stream>>>(h, tx1, src, dst, nrm, E);
    hipMemsetAsync(tx2, 0, HS * 4, stream);
    k_prop<<<prop_blocks, 256, 0, stream>>>(tx1, tx2, src, dst, nrm, E);
    k_fixup<<<(unsigned)((total + 255) / 256), 256, 0, stream>>>(tx2, h, total);
    k_cheb_gemm<<<gemm_grid, 256, 0, stream>>>(h, tx1, tx2,
        cheb_w + (size_t)l * KCHEB * HDIM * HDIM, cheb_b + l * HDIM, accP);
    k_tconv_gemm<<<gemm_grid, 256, 0, stream>>>(accP,
        conv_w + (size_t)l * HDIM * HDIM * 3, conv_b + l * HDIM, tx1);
    k_ln_relu<<<row_blocks, 256, 0, stream>>>(tx1, h, ln_g + l * HDIM, ln_b + l * HDIM);
  }

  k_out_proj<<<row_blocks, 256, 0, stream>>>(h, out_w, out_b, (float*)d_out);
}